// YOLO_36077725286698
// MI455X (gfx1250) — compile-verified
//
#include <hip/hip_runtime.h>
#include <stdint.h>
#include <math.h>

// Problem constants (fixed by the reference).
#define BB 4
#define NN 8400
#define CC 81
#define MM 16384                 // pow2 padding for bitonic sort (per batch)
#define TILE 2048                // LDS-resident sort tile
#define LTH 1024                 // threads for local sort kernels (TILE/2 pairs)
#define CH 2048                  // NMS chunk (boxes staged in LDS, 32 KB)
#define NCH 5                    // ceil(8400/2048)
#define NS (CH * NCH)            // 10240 sorted slots (>= NN; rest are pads)
#define SPLIT 8                  // cross-pass slices per (batch, c1) chunk
#define CTH 256                  // cross-kernel block size (== CH/SPLIT)
#define IOU_THR 0.5f
#define BBOX_CLIP 4.135166556742356f   // log(1000/16)
#define NEGBIG -1e9f

// ---------------------------------------------------------------------------
// Kernel 1: class max/argmax + box decode + sort-key emission.
// ~11 MB of input -> ~0.5us at 23.3 TB/s; one thread per box.
// ---------------------------------------------------------------------------
__global__ void yolo_init_kernel(const float* __restrict__ labels,
                                 const float* __restrict__ deltas,
                                 const float* __restrict__ anchors,
                                 float* __restrict__ out,
                                 float* __restrict__ keys,
                                 int* __restrict__ sidx) {
    int gid = blockIdx.x * blockDim.x + threadIdx.x;
    if (gid >= BB * MM) return;
    int b = gid / MM;
    int t = gid - b * MM;

    if (t >= NN) {               // sort padding: sinks below every real entry
        keys[gid] = -INFINITY;
        sidx[gid] = t;
        return;
    }

    const float* pl = labels + (size_t)(b * NN + t) * CC;
    __builtin_prefetch(pl, 0, 0);            // -> global_prefetch_b8
    float m = pl[0];
    int am = 0;
    for (int c = 1; c < CC; ++c) {           // first-max semantics (jnp.argmax)
        float v = pl[c];
        if (v > m) { m = v; am = c; }
    }
    bool valid = (am != 0);
    float score = valid ? m : NEGBIG;

    const float* a = anchors + (size_t)(b * NN + t) * 4;
    const float* d = deltas  + (size_t)(b * NN + t) * 4;
    float w  = a[2] - a[0];
    float h  = a[3] - a[1];
    float cx = a[0] + 0.5f * w;
    float cy = a[1] + 0.5f * h;
    float dw = fminf(d[2], BBOX_CLIP);
    float dh = fminf(d[3], BBOX_CLIP);
    float pcx = d[0] * w + cx;
    float pcy = d[1] * h + cy;
    float pw  = expf(dw) * w;
    float ph  = expf(dh) * h;

    float* ob = out + (size_t)(b * NN + t) * 4;
    ob[0] = pcx - 0.5f * pw;
    ob[1] = pcy - 0.5f * ph;
    ob[2] = pcx + 0.5f * pw;
    ob[3] = pcy + 0.5f * ph;

    float* out_scores  = out + (size_t)BB * NN * 4;
    float* out_classes = out + (size_t)BB * NN * 5;
    out_scores[b * NN + t]  = score;
    out_classes[b * NN + t] = (float)am;

    keys[gid] = score;           // sort key == masked score (matches reference)
    sidx[gid] = t;
}

// ---------------------------------------------------------------------------
// Sort kernels (descending per batch). Unstable ordering is exact: ties exist
// only among background(-1e9)/pad(-INF) entries, which never suppress and are
// always keep=false. Direction bit MUST come from the within-batch index.
// ---------------------------------------------------------------------------
__device__ __forceinline__ void cmpxchg_desc(float* sk, int* si,
                                             int i, int l, bool up) {
    float ki = sk[i], kl = sk[l];
    bool sw = up ? (ki < kl) : (ki > kl);    // descending network
    if (sw) {
        sk[i] = kl; sk[l] = ki;
        int ti = si[i]; si[i] = si[l]; si[l] = ti;
    }
}

// All stages k = 2..TILE fully inside LDS: replaces 66 global launches with 1.
__global__ __launch_bounds__(LTH)
void bitonic_local_sort(float* __restrict__ keys, int* __restrict__ sidx) {
    __shared__ float sk[TILE];
    __shared__ int   si[TILE];
    int tile0 = blockIdx.x * TILE;           // tiles never cross batch bounds
    int tid = threadIdx.x;
    for (int e = tid; e < TILE; e += LTH) { sk[e] = keys[tile0 + e];
                                            si[e] = sidx[tile0 + e]; }
    __syncthreads();
    for (int k = 2; k <= TILE; k <<= 1) {
        for (int j = k >> 1; j >= 1; j >>= 1) {
            int i = ((tid & ~(j - 1)) << 1) | (tid & (j - 1));
            int ib = (tile0 + i) & (MM - 1); // index within batch
            cmpxchg_desc(sk, si, i, i + j, (ib & k) == 0);
            __syncthreads();
        }
    }
    for (int e = tid; e < TILE; e += LTH) { keys[tile0 + e] = sk[e];
                                            sidx[tile0 + e] = si[e]; }
}

// One cross-tile compare-exchange step (j >= TILE) in global memory.
__global__ void bitonic_step_kernel(float* __restrict__ keys,
                                    int* __restrict__ sidx,
                                    int k, int j) {
    int gid = blockIdx.x * blockDim.x + threadIdx.x;
    if (gid >= BB * MM) return;
    int b = gid / MM;
    int i = gid - b * MM;
    int l = i ^ j;
    if (l <= i) return;
    int base = b * MM;
    float ki = keys[base + i];
    float kl = keys[base + l];
    bool up = ((i & k) == 0);
    bool sw = up ? (ki < kl) : (ki > kl);
    if (sw) {
        keys[base + i] = kl; keys[base + l] = ki;
        int ti = sidx[base + i];
        sidx[base + i] = sidx[base + l];
        sidx[base + l] = ti;
    }
}

// For a given k > TILE: all remaining steps j = TILE/2..1 inside LDS.
__global__ __launch_bounds__(LTH)
void bitonic_local_merge(float* __restrict__ keys, int* __restrict__ sidx,
                         int k) {
    __shared__ float sk[TILE];
    __shared__ int   si[TILE];
    int tile0 = blockIdx.x * TILE;
    int tid = threadIdx.x;
    for (int e = tid; e < TILE; e += LTH) { sk[e] = keys[tile0 + e];
                                            si[e] = sidx[tile0 + e]; }
    __syncthreads();
    for (int j = TILE >> 1; j >= 1; j >>= 1) {
        int i = ((tid & ~(j - 1)) << 1) | (tid & (j - 1));
        int ib = (tile0 + i) & (MM - 1);     // within-batch index (k may == MM)
        cmpxchg_desc(sk, si, i, i + j, (ib & k) == 0);
        __syncthreads();
    }
    for (int e = tid; e < TILE; e += LTH) { keys[tile0 + e] = sk[e];
                                            sidx[tile0 + e] = si[e]; }
}

// ---------------------------------------------------------------------------
// Kernel: initialize the global keep-byte array from sorted keys.
// keep == foreground validity (valid scores >> -1e8 > background/pad keys).
// ---------------------------------------------------------------------------
__global__ void kp_init_kernel(const float* __restrict__ keys,
                               unsigned char* __restrict__ kp) {
    int gid = blockIdx.x * blockDim.x + threadIdx.x;
    if (gid >= BB * NS) return;
    int b = gid / NS;
    int j = gid - b * NS;
    kp[gid] = (keys[b * MM + j] > -1e8f) ? 1 : 0;
}

// ---------------------------------------------------------------------------
// NMS helpers.
// ---------------------------------------------------------------------------
__device__ __forceinline__ float iou_ref(float4 a, float areaA, float4 bx) {
    float areaB = (bx.z - bx.x) * (bx.w - bx.y);
    float ltx = fmaxf(a.x, bx.x), lty = fmaxf(a.y, bx.y);
    float rbx = fminf(a.z, bx.z), rby = fminf(a.w, bx.w);
    float iw = fmaxf(rbx - ltx, 0.0f);
    float ih = fmaxf(rby - lty, 0.0f);
    float inter = iw * ih;
    return inter / (areaA + areaB - inter);   // suppressor area > 0 => safe
}

// Stage one chunk of sorted boxes into LDS via the gfx1250 async DMA path
// (ASYNCcnt-tracked), gathered through the sorted-index array. Dead entries
// (already suppressed / background / pad) are zero-filled and never read.
__device__ __forceinline__ void stage_chunk(float4* dst,                 // LDS
                                            const float* __restrict__ boxes_b,
                                            const int* __restrict__ idx_b,
                                            const unsigned char* __restrict__ kp_b,
                                            int c, int tid, int nthr) {
    int base = c * CH;
    for (int ii = tid; ii < CH; ii += nthr) {
        int g = base + ii;
        int id = idx_b[g];
        if (id < NN && kp_b[g]) {
            // Flat pointer to LDS: low 32 bits == LDS byte address.
            unsigned ldsoff = (unsigned)(uintptr_t)(&dst[ii]);
            const float* src = boxes_b + (size_t)id * 4;
            asm volatile("global_load_async_to_lds_b128 %0, %1, off"
                         :: "v"(ldsoff), "v"(src) : "memory");
        } else {
            dst[ii] = make_float4(0.f, 0.f, 0.f, 0.f);
        }
    }
    asm volatile("s_wait_asynccnt 0x0" ::: "memory");    // DMA -> LDS visible
    __syncthreads();
}

// ---------------------------------------------------------------------------
// Intra-chunk exact greedy NMS for chunk c0. One WG (32 wave32s) per batch.
// The truly serial part runs inside a SINGLE wave over 32-box sub-chunks with
// a register alive-mask + ballot (lockstep, no barriers); the chunk tail is
// then suppressed in parallel by all 1024 threads.
// ---------------------------------------------------------------------------
__global__ __launch_bounds__(1024)
void nms_intra_kernel(const float* __restrict__ boxes,
                      const int* __restrict__ sidx,
                      unsigned char* __restrict__ kp,
                      int c0) {
    __shared__ float4 bufA[CH];          // 32 KB
    __shared__ unsigned char ck[CH];     //  2 KB
    int b = blockIdx.x;
    int tid = threadIdx.x;
    const int nthr = 1024;
    const float* boxes_b = boxes + (size_t)b * NN * 4;
    const int*   idx_b   = sidx + (size_t)b * MM;
    unsigned char* kp_b  = kp + (size_t)b * NS;
    int base0 = c0 * CH;

    for (int ii = tid; ii < CH; ii += nthr) ck[ii] = kp_b[base0 + ii];
    stage_chunk(bufA, boxes_b, idx_b, kp_b, c0, tid, nthr);  // incl. barrier

    for (int s = 0; s < CH / 32; ++s) {
        int lb = s * 32;
        // --- wave-serial greedy over the 32-box sub-chunk (wave 0 only) ---
        if (tid < 32) {
            bool alive = ck[lb + tid] != 0;
            float4 bm = bufA[lb + tid];
            for (int i = 0; i < 32; ++i) {
                unsigned msk = __builtin_amdgcn_ballot_w32(alive);
                if ((msk >> i) & 1u) {                  // keep[i] is current
                    float4 bi = bufA[lb + i];           // LDS broadcast
                    float ai = (bi.z - bi.x) * (bi.w - bi.y);
                    if (tid > i && alive && iou_ref(bi, ai, bm) > IOU_THR)
                        alive = false;
                }
            }
            ck[lb + tid] = alive ? 1 : 0;
        }
        __syncthreads();
        // --- finalized sub-chunk suppresses the rest of the chunk (parallel) ---
        for (int jj = lb + 32 + tid; jj < CH; jj += nthr) {
            if (!ck[jj]) continue;
            float4 bj = bufA[jj];
            bool dead = false;
            for (int i = 0; i < 32 && !dead; ++i) {
                if (ck[lb + i]) {
                    float4 bi = bufA[lb + i];
                    float ai = (bi.z - bi.x) * (bi.w - bi.y);
                    dead = iou_ref(bi, ai, bj) > IOU_THR;
                }
            }
            if (dead) ck[jj] = 0;
        }
        __syncthreads();
    }

    for (int ii = tid; ii < CH; ii += nthr) kp_b[base0 + ii] = ck[ii];
}

// ---------------------------------------------------------------------------
// Cross-chunk suppression c0 -> all c1 > c0. Exact because keep bits of chunk
// c0 are final before any later-chunk entry is decided, so this phase is
// embarrassingly parallel: grid = B * (NCH-1-c0) * SPLIT WGs, each owning a
// disjoint 256-box slice of some chunk c1 (no write races). Each WG stages
// chunk c0 into LDS via async DMA.
// ---------------------------------------------------------------------------
__global__ __launch_bounds__(CTH)
void nms_cross_kernel(const float* __restrict__ boxes,
                      const int* __restrict__ sidx,
                      unsigned char* __restrict__ kp,
                      int c0) {
    __shared__ float4 bufA[CH];
    __shared__ unsigned char ck[CH];
    int nrem = NCH - 1 - c0;
    int wg = blockIdx.x;
    int slice = wg % SPLIT;
    int t = wg / SPLIT;
    int c1 = c0 + 1 + (t % nrem);
    int b = t / nrem;
    int tid = threadIdx.x;
    const float* boxes_b = boxes + (size_t)b * NN * 4;
    const int*   idx_b   = sidx + (size_t)b * MM;
    unsigned char* kp_b  = kp + (size_t)b * NS;
    int base0 = c0 * CH;

    for (int ii = tid; ii < CH; ii += CTH) ck[ii] = kp_b[base0 + ii];
    stage_chunk(bufA, boxes_b, idx_b, kp_b, c0, tid, CTH);   // incl. barrier

    int jj = slice * CTH + tid;                  // CH/SPLIT == CTH
    int gj = c1 * CH + jj;
    if (!kp_b[gj]) return;                       // dead/background/pad
    int id = idx_b[gj];                          // keep==1 => id < NN
    float4 bj = ((const float4*)boxes_b)[id];
    bool dead = false;
    for (int i = 0; i < CH && !dead; ++i) {
        if (ck[i]) {
            float4 bi = bufA[i];
            float ai = (bi.z - bi.x) * (bi.w - bi.y);
            dead = iou_ref(bi, ai, bj) > IOU_THR;
        }
    }
    if (dead) kp_b[gj] = 0;
}

// ---------------------------------------------------------------------------
// Scatter keep mask back to original (unsorted) box order. All NN real
// entries occupy sorted slots [0, NN) since pads carry -INF keys.
// ---------------------------------------------------------------------------
__global__ void scatter_keep_kernel(const int* __restrict__ sidx,
                                    const unsigned char* __restrict__ kp,
                                    float* __restrict__ out_keep) {
    int gid = blockIdx.x * blockDim.x + threadIdx.x;
    if (gid >= BB * NS) return;
    int b = gid / NS;
    int j = gid - b * NS;
    int id = sidx[b * MM + j];
    if (id < NN)
        out_keep[(size_t)b * NN + id] = kp[gid] ? 1.0f : 0.0f;
}

// ---------------------------------------------------------------------------
// Host side. Workspace: keys (256KB) + sidx (256KB) + keep bytes (40KB).
// Launches: 1 init + 10 sort (1 local + 6 global steps + 3 local merges)
//         + 1 kp-init + 5 intra + 4 cross + 1 scatter  == 22 graph nodes
// (vs 117 before: dispatch latency was the dominant serial term).
// ---------------------------------------------------------------------------
extern "C" void kernel_launch(void* const* d_in, const int* in_sizes, int n_in,
                              void* d_out, int out_size, void* d_ws, size_t ws_size,
                              hipStream_t stream) {
    const float* labels  = (const float*)d_in[0];
    const float* deltas  = (const float*)d_in[1];
    const float* anchors = (const float*)d_in[2];
    float* out = (float*)d_out;

    float* keys = (float*)d_ws;
    int*   sidx = (int*)((char*)d_ws + (size_t)BB * MM * sizeof(float));
    unsigned char* kp = (unsigned char*)((char*)d_ws +
                        (size_t)BB * MM * (sizeof(float) + sizeof(int)));

    const int total  = BB * MM;          // 65536
    const int blocks = total / 256;
    const int ntiles = total / TILE;     // 32

    yolo_init_kernel<<<blocks, 256, 0, stream>>>(labels, deltas, anchors,
                                                 out, keys, sidx);

    // Fused bitonic sort: k <= TILE in LDS, cross-tile steps global, then
    // per-k LDS merge of the remaining j <= TILE/2 steps.
    bitonic_local_sort<<<ntiles, LTH, 0, stream>>>(keys, sidx);
    for (int k = TILE * 2; k <= MM; k <<= 1) {
        for (int j = k >> 1; j >= TILE; j >>= 1)
            bitonic_step_kernel<<<blocks, 256, 0, stream>>>(keys, sidx, k, j);
        bitonic_local_merge<<<ntiles, LTH, 0, stream>>>(keys, sidx, k);
    }

    kp_init_kernel<<<(BB * NS + 255) / 256, 256, 0, stream>>>(keys, kp);

    for (int c0 = 0; c0 < NCH; ++c0) {
        nms_intra_kernel<<<BB, 1024, 0, stream>>>(out, sidx, kp, c0);
        int nrem = NCH - 1 - c0;
        if (nrem > 0)
            nms_cross_kernel<<<BB * nrem * SPLIT, CTH, 0, stream>>>(out, sidx,
                                                                    kp, c0);
    }

    float* out_keep = out + (size_t)BB * NN * 6;   // after boxes|scores|classes
    scatter_keep_kernel<<<(BB * NS + 255) / 256, 256, 0, stream>>>(sidx, kp,
                                                                   out_keep);
}